// PointerNet_9431748182358
// MI455X (gfx1250) — compile-verified
//
#include <hip/hip_runtime.h>
#include <math.h>

// Problem constants (from reference)
#define BB 32
#define SS 1024
#define EE 512
#define TT 128
#define QQ 256
#define HH 4
#define FF 1024   // H*Q
#define BS 32768  // B*S

typedef __attribute__((ext_vector_type(2))) float v2f;
typedef __attribute__((ext_vector_type(8))) float v8f;

__device__ __forceinline__ v8f wmma_f32(v2f a, v2f b, v8f c) {
    // D = A(16x4) * B(4x16) + C, all f32, f32 accumulate
    return __builtin_amdgcn_wmma_f32_16x16x4_f32(false, a, false, b, (short)0, c, false, false);
}

// ---------------------------------------------------------------------------
// Kernel 1: options_scores = query . W_opt^T ; best = argmax_h (first wins)
// ---------------------------------------------------------------------------
__global__ __launch_bounds__(256) void head_argmax(const float* __restrict__ qv,
                                                   const float* __restrict__ Wopt,
                                                   int* __restrict__ best) {
    int idx = blockIdx.x * blockDim.x + threadIdx.x;  // t*B + b
    if (idx >= TT * BB) return;
    const float* q = qv + (size_t)idx * QQ;
    float s0 = 0.f, s1 = 0.f, s2 = 0.f, s3 = 0.f;
    for (int e = 0; e < QQ; ++e) {
        float x = q[e];
        s0 += x * Wopt[0 * QQ + e];
        s1 += x * Wopt[1 * QQ + e];
        s2 += x * Wopt[2 * QQ + e];
        s3 += x * Wopt[3 * QQ + e];
    }
    int bh = 0; float bv = s0;
    if (s1 > bv) { bv = s1; bh = 1; }
    if (s2 > bv) { bv = s2; bh = 2; }
    if (s3 > bv) { bv = s3; bh = 3; }
    best[idx] = bh;
}

// ---------------------------------------------------------------------------
// Kernel 2: proj = leaky_relu(src(32768x512) @ W_src^T(512x1024)) -> (32768x1024)
// 8 waves/block, block tile 256(M) x 128(N), wave tile 64x64 via 4x4 f32 WMMA frags
// ---------------------------------------------------------------------------
__global__ __launch_bounds__(256) void proj_gemm(const float* __restrict__ src,
                                                 const float* __restrict__ Wsrc,
                                                 float* __restrict__ proj) {
    const int lane  = threadIdx.x & 31;
    const int wave  = threadIdx.x >> 5;       // 0..7
    const int waveM = (wave >> 1) * 64;       // 4 waves along M
    const int waveN = (wave & 1) * 64;        // 2 waves along N
    const int blockM = blockIdx.x * 256;
    const int blockN = blockIdx.y * 128;
    const int lm = lane & 15;
    const int kh = (lane >> 4) * 2;           // lanes 16-31 hold K=2,3 of the 16x4 tile

    v8f acc[4][4] = {};
    const int rowA0 = blockM + waveM + lm;    // src row (+ mf*16)
    const int rowB0 = blockN + waveN + lm;    // W_src row == output col (+ nf*16)

    for (int k0 = 0; k0 < EE; k0 += 4) {
        const int k = k0 + kh;
        v2f a[4], bm[4];
#pragma unroll
        for (int mf = 0; mf < 4; ++mf)
            a[mf] = *reinterpret_cast<const v2f*>(src + (size_t)(rowA0 + mf * 16) * EE + k);
#pragma unroll
        for (int nf = 0; nf < 4; ++nf)
            bm[nf] = *reinterpret_cast<const v2f*>(Wsrc + (size_t)(rowB0 + nf * 16) * EE + k);
#pragma unroll
        for (int mf = 0; mf < 4; ++mf)
#pragma unroll
            for (int nf = 0; nf < 4; ++nf)
                acc[mf][nf] = wmma_f32(a[mf], bm[nf], acc[mf][nf]);
    }

    const int hi = lane >> 4;
#pragma unroll
    for (int mf = 0; mf < 4; ++mf) {
#pragma unroll
        for (int nf = 0; nf < 4; ++nf) {
#pragma unroll
            for (int r = 0; r < 8; ++r) {
                int row = blockM + waveM + mf * 16 + r + 8 * hi;
                int col = blockN + waveN + nf * 16 + lm;
                float v = acc[mf][nf][r];
                v = v > 0.f ? v : 0.01f * v;          // leaky_relu, slope 0.01
                proj[(size_t)row * FF + col] = v;
            }
        }
    }
}

// ---------------------------------------------------------------------------
// Kernel 3 (fused): for each (b, 16-t chunk): scores for all 4 heads computed
// sequentially with f32 WMMA, chosen head selected per row via best[t,b],
// mask -> -inf, then row softmax across the block (each block owns full S rows).
// 8 waves, each wave covers 128 of the 1024 source positions (8 nf frags).
// Output written once: (T,B,S) probabilities.
// ---------------------------------------------------------------------------
__global__ __launch_bounds__(256) void score_softmax(const float* __restrict__ qv,
                                                     const float* __restrict__ proj,
                                                     const int* __restrict__ best,
                                                     const unsigned char* __restrict__ mask,
                                                     float* __restrict__ out) {
    const int b     = blockIdx.y;
    const int tBase = blockIdx.x * 16;
    const int lane = threadIdx.x & 31;
    const int wave = threadIdx.x >> 5;
    const int lm = lane & 15;
    const int hi = lane >> 4;
    const int kh = hi * 2;
    const int colBase = wave * 128;           // this wave's S-column window

    // chosen head for each of this lane-half's 8 rows (rows r + 8*hi)
    int bestv[8];
#pragma unroll
    for (int r = 0; r < 8; ++r)
        bestv[r] = best[(tBase + r + 8 * hi) * BB + b];

    float selv[8][8];                         // [nf][r] chosen-head scores
#pragma unroll
    for (int nf = 0; nf < 8; ++nf)
#pragma unroll
        for (int r = 0; r < 8; ++r) selv[nf][r] = 0.f;

    const float* aptr = qv + ((size_t)(tBase + lm) * BB + b) * QQ;

    for (int h = 0; h < HH; ++h) {            // heads sequentially (low VGPR)
        v8f acc[8] = {};
        const float* bptr = proj + ((size_t)b * SS + colBase + lm) * FF + h * QQ;
        for (int k0 = 0; k0 < QQ; k0 += 4) {
            const int k = k0 + kh;
            v2f a = *reinterpret_cast<const v2f*>(aptr + k);
            v2f bf[8];
#pragma unroll
            for (int nf = 0; nf < 8; ++nf)
                bf[nf] = *reinterpret_cast<const v2f*>(bptr + (size_t)nf * 16 * FF + k);
#pragma unroll
            for (int nf = 0; nf < 8; ++nf)
                acc[nf] = wmma_f32(a, bf[nf], acc[nf]);
        }
#pragma unroll
        for (int nf = 0; nf < 8; ++nf)
#pragma unroll
            for (int r = 0; r < 8; ++r)
                if (bestv[r] == h) selv[nf][r] = acc[nf][r];
    }

    // mask -> -inf (mask depends only on column, same for all rows)
#pragma unroll
    for (int nf = 0; nf < 8; ++nf) {
        unsigned char mb = mask[(size_t)b * SS + colBase + nf * 16 + lm];
        if (mb) {
#pragma unroll
            for (int r = 0; r < 8; ++r) selv[nf][r] = -INFINITY;
        }
    }

    // ---- row softmax across the whole block ----
    __shared__ float smax[8][16];
    __shared__ float ssum[8][16];

    // per-wave partial row max (reduce over nf frags + 16 lanes of this half)
    float pm[8];
#pragma unroll
    for (int r = 0; r < 8; ++r) {
        float m = selv[0][r];
#pragma unroll
        for (int nf = 1; nf < 8; ++nf) m = fmaxf(m, selv[nf][r]);
#pragma unroll
        for (int off = 1; off <= 8; off <<= 1) m = fmaxf(m, __shfl_xor(m, off, 32));
        pm[r] = m;
    }
    if (lm == 0) {
#pragma unroll
        for (int r = 0; r < 8; ++r) smax[wave][r + 8 * hi] = pm[r];
    }
    __syncthreads();
    float rowmax[8];
#pragma unroll
    for (int r = 0; r < 8; ++r) {
        float m = smax[0][r + 8 * hi];
#pragma unroll
        for (int w = 1; w < 8; ++w) m = fmaxf(m, smax[w][r + 8 * hi]);
        rowmax[r] = m;
    }

    // exp + per-wave partial row sum
    float ps[8];
#pragma unroll
    for (int r = 0; r < 8; ++r) {
        float s = 0.f;
#pragma unroll
        for (int nf = 0; nf < 8; ++nf) {
            selv[nf][r] = expf(selv[nf][r] - rowmax[r]);  // exp(-inf)=0 for masked
            s += selv[nf][r];
        }
#pragma unroll
        for (int off = 1; off <= 8; off <<= 1) s += __shfl_xor(s, off, 32);
        ps[r] = s;
    }
    if (lm == 0) {
#pragma unroll
        for (int r = 0; r < 8; ++r) ssum[wave][r + 8 * hi] = ps[r];
    }
    __syncthreads();

    // normalize and write final probabilities (single store of the output)
#pragma unroll
    for (int r = 0; r < 8; ++r) {
        float s = ssum[0][r + 8 * hi];
#pragma unroll
        for (int w = 1; w < 8; ++w) s += ssum[w][r + 8 * hi];
        float inv = 1.0f / s;
        const int t = tBase + r + 8 * hi;
#pragma unroll
        for (int nf = 0; nf < 8; ++nf)
            out[((size_t)t * BB + b) * SS + colBase + nf * 16 + lm] = selv[nf][r] * inv;
    }
}

// ---------------------------------------------------------------------------
extern "C" void kernel_launch(void* const* d_in, const int* in_sizes, int n_in,
                              void* d_out, int out_size, void* d_ws, size_t ws_size,
                              hipStream_t stream) {
    const float*         src  = (const float*)d_in[0];         // (B,S,E)
    const unsigned char* mask = (const unsigned char*)d_in[1]; // (B,S) bool (1 byte)
    const float*         qv   = (const float*)d_in[2];         // (T,B,Q)
    const float*         Wsrc = (const float*)d_in[3];         // (H*Q, E)
    const float*         Wopt = (const float*)d_in[4];         // (H, Q)
    float* out = (float*)d_out;                                // (T,B,S)

    float* proj = (float*)d_ws;                                // 32768*1024 f32 = 128 MiB
    int*   best = (int*)((char*)d_ws + (size_t)BS * FF * sizeof(float)); // 4096 ints

    head_argmax  <<<dim3((TT * BB + 255) / 256), dim3(256), 0, stream>>>(qv, Wopt, best);
    proj_gemm    <<<dim3(BS / 256, FF / 128),    dim3(256), 0, stream>>>(src, Wsrc, proj);
    score_softmax<<<dim3(TT / 16, BB),           dim3(256), 0, stream>>>(qv, proj, best, mask, out);
}